// Qwen2VisionTransformerPretrainedModel_VisionZip_44581760532858
// MI455X (gfx1250) — compile-verified
//
#include <hip/hip_runtime.h>
#include <math.h>

// ---------------- problem constants (B=1) ----------------
constexpr int N    = 2304;          // tokens
constexpr int D    = 1280;          // hidden dim
constexpr int H    = 16;            // heads
constexpr int Dm   = 80;            // metric dim
constexpr int NM1  = N - 1;         // 2303 (cls attention targets)
constexpr int DOM  = 971;           // dominant tokens (top-k)
constexpr int K1   = DOM + 1;       // 972 kept (incl. CLS)
constexpr int NF   = N - K1;        // 1332 filtered
constexpr int CTX  = 180;           // contextual targets
constexpr int STEP = 7;             // 1332 // 180
constexpr int ROWS = K1 + CTX;      // 1152 final tokens
constexpr int HS   = D * 4;         // 5120 (merge 4 tokens)
constexpr int XROWS = ROWS / 4;     // 288 MLP rows
constexpr int N2   = 3584;          // output dim

typedef __attribute__((ext_vector_type(2))) float v2f;
typedef __attribute__((ext_vector_type(8))) float v8f;

// ---- 1. cls_attention_sum[j] = sum_h attn[h][0][1+j] ----
__global__ void k_cls(const float* __restrict__ attn, float* __restrict__ cls) {
    int j = blockIdx.x * blockDim.x + threadIdx.x;
    if (j >= NM1) return;
    float s = 0.f;
    #pragma unroll
    for (int h = 0; h < H; ++h)
        s += attn[(size_t)h * N * N + (1 + j)];
    cls[j] = s;
}

// ---- 2. exact stable top-k via rank counting (matches jax.lax.top_k ties) ----
__global__ void k_rank(const float* __restrict__ cls, int* __restrict__ keep) {
    __shared__ float sv[NM1];
    for (int i = threadIdx.x; i < NM1; i += blockDim.x) sv[i] = cls[i];
    __syncthreads();
    int j = blockIdx.x * blockDim.x + threadIdx.x;
    if (j == 0) keep[0] = 1;                    // CLS always kept
    if (j >= NM1) return;
    float v = sv[j];
    int rank = 0;
    for (int i = 0; i < NM1; ++i) {
        float u = sv[i];
        rank += (u > v) || ((u == v) && (i < j));
    }
    keep[1 + j] = (rank < DOM) ? 1 : 0;
}

// ---- 3. stable partition of indices: kept first, then filtered ----
__global__ void k_positions(const int* __restrict__ keep,
                            int* __restrict__ dom_pos, int* __restrict__ filt_pos) {
    if (blockIdx.x != 0 || threadIdx.x != 0) return;
    int nd = 0, nf = 0;
    for (int i = 0; i < N; ++i) {
        if (keep[i]) dom_pos[nd++] = i;
        else         filt_pos[nf++] = i;
    }
}

// ---- 4. gather + L2-normalize metric rows ----
__global__ void k_metric(const float* __restrict__ metric, const int* __restrict__ filt_pos,
                         float* __restrict__ mnorm) {
    int f = blockIdx.x;                 // 0..NF-1
    int t = threadIdx.x;                // 128 threads
    __shared__ float red[128];
    float val = 0.f, sq = 0.f;
    if (t < Dm) {
        val = metric[(size_t)filt_pos[f] * Dm + t];
        sq = val * val;
    }
    red[t] = sq;
    __syncthreads();
    for (int s = 64; s > 0; s >>= 1) {
        if (t < s) red[t] += red[t + s];
        __syncthreads();
    }
    float inv = 1.0f / sqrtf(red[0]);
    if (t < Dm) mnorm[(size_t)f * Dm + t] = val * inv;
}

// ---- 5. argmax cosine similarity against the 180 targets (filtered idx 7c) ----
__global__ void k_assign(const float* __restrict__ mnorm, int* __restrict__ assignv) {
    int m = blockIdx.x * blockDim.x + threadIdx.x;
    if (m >= NF) return;
    bool is_target = ((m % STEP) == 0) && (m < CTX * STEP);
    if (is_target) { assignv[m] = -1; return; }
    const float* mv = mnorm + (size_t)m * Dm;
    float best = -3.402823e38f; int bi = 0;
    for (int c = 0; c < CTX; ++c) {
        const float* tv = mnorm + (size_t)(c * STEP) * Dm;
        float d = 0.f;
        #pragma unroll 8
        for (int e = 0; e < Dm; ++e) d += mv[e] * tv[e];
        if (d > best) { best = d; bi = c; }     // first max wins (argmax semantics)
    }
    assignv[m] = bi;
}

// ---- 6. per-target mean of assigned hidden rows (atomic-free, deterministic) ----
__global__ void k_aggregate(const float* __restrict__ hs, const int* __restrict__ filt_pos,
                            const int* __restrict__ assignv, float* __restrict__ agg) {
    int c = blockIdx.x;                 // 0..CTX-1
    int t = threadIdx.x;                // 256 threads, 5 cols each
    float sum[5] = {0.f, 0.f, 0.f, 0.f, 0.f};
    int cnt = 0;
    for (int m = 0; m < NF; ++m) {
        if (assignv[m] == c) {
            ++cnt;
            const float* row = hs + (size_t)filt_pos[m] * D;
            #pragma unroll
            for (int j = 0; j < 5; ++j) sum[j] += row[t + j * 256];
        }
    }
    float inv = 1.0f / (float)(cnt > 0 ? cnt : 1);
    #pragma unroll
    for (int j = 0; j < 5; ++j) agg[(size_t)c * D + t + j * 256] = sum[j] * inv;
}

// ---- 7. build final tokens + LayerNorm -> X (viewed as XROWS x HS) ----
__global__ void k_final_ln(const float* __restrict__ hs, const int* __restrict__ dom_pos,
                           const float* __restrict__ agg,
                           const float* __restrict__ ln_w, const float* __restrict__ ln_b,
                           float* __restrict__ X) {
    int r = blockIdx.x;                 // 0..ROWS-1
    int t = threadIdx.x;                // 256 threads, 5 cols each
    float v[5];
    if (r < K1) {
        const float* src = hs + (size_t)dom_pos[r] * D;
        #pragma unroll
        for (int j = 0; j < 5; ++j) v[j] = src[t + j * 256];
    } else {
        int c = r - K1;
        // reference quirk: target_hidden = hidden_states[:, tgt] (unfiltered index 7c)
        const float* src = hs + (size_t)(c * STEP) * D;
        #pragma unroll
        for (int j = 0; j < 5; ++j) v[j] = src[t + j * 256] + agg[(size_t)c * D + t + j * 256];
    }
    __shared__ float s1[256], s2[256];
    float a = 0.f, b = 0.f;
    #pragma unroll
    for (int j = 0; j < 5; ++j) { a += v[j]; b += v[j] * v[j]; }
    s1[t] = a; s2[t] = b;
    __syncthreads();
    for (int s = 128; s > 0; s >>= 1) {
        if (t < s) { s1[t] += s1[t + s]; s2[t] += s2[t + s]; }
        __syncthreads();
    }
    float mean = s1[0] * (1.0f / D);
    float var  = s2[0] * (1.0f / D) - mean * mean;
    float inv  = rsqrtf(var + 1e-5f);
    #pragma unroll
    for (int j = 0; j < 5; ++j) {
        int d = t + j * 256;
        X[(size_t)r * D + d] = (v[j] - mean) * inv * ln_w[d] + ln_b[d];
    }
}

// ---- 8. WMMA fp32 GEMM: C[m][n] = sum_k A[m][k]*W[n][k] + bias[n] (opt. exact GELU)
// Block = 4 waves; each wave owns a 16x64 output strip (4 f32 accumulators).
// Depth-4 software pipeline with UNCONDITIONAL steady-state prefetch:
// main loop runs k0 <= K-32, so prefetch indices (k0+16 .. k0+31) are always
// in range -> no clamping math, the compiler keeps pointer-increment + immediate
// offset addressing, and up to 20 fragment loads (~5 KB/wave) stay in flight
// behind the WMMAs. Load-free epilogue consumes the last 4 stages.
template <bool GELU>
__global__ void gemm_wmma_f32(const float* __restrict__ A, const float* __restrict__ W,
                              const float* __restrict__ bias, float* __restrict__ C,
                              int K, int Ncols) {
    const int lane  = threadIdx.x & 31;
    const int wave  = threadIdx.x >> 5;                 // 0..3
    const int mBase = blockIdx.x * 16;
    const int nBase = blockIdx.y * 256 + wave * 64;
    const int l16   = lane & 15;
    const int kOff  = (lane >> 4) * 2;                  // lanes 0-15 -> k+0,1 ; 16-31 -> k+2,3

    const float* aRow = A + (size_t)(mBase + l16) * K;
    const float* wr0  = W + (size_t)(nBase + 0 * 16 + l16) * K;
    const float* wr1  = W + (size_t)(nBase + 1 * 16 + l16) * K;
    const float* wr2  = W + (size_t)(nBase + 2 * 16 + l16) * K;
    const float* wr3  = W + (size_t)(nBase + 3 * 16 + l16) * K;

    auto ld = [&](const float* p, int ka) -> v2f { return v2f{ p[ka], p[ka + 1] }; };

    v8f acc0 = {}, acc1 = {}, acc2 = {}, acc3 = {};

    // preload 4 pipeline stages: k = 0, 4, 8, 12
    v2f a[4], b0[4], b1[4], b2[4], b3[4];
    #pragma unroll
    for (int s = 0; s < 4; ++s) {
        int ka = 4 * s + kOff;
        a[s]  = ld(aRow, ka);
        b0[s] = ld(wr0, ka);
        b1[s] = ld(wr1, ka);
        b2[s] = ld(wr2, ka);
        b3[s] = ld(wr3, ka);
    }

    // steady state: compute stage s at k0+4s, prefetch k0+16+4s (always in range)
    for (int k0 = 0; k0 + 16 < K; k0 += 16) {
        #pragma unroll
        for (int s = 0; s < 4; ++s) {
            acc0 = __builtin_amdgcn_wmma_f32_16x16x4_f32(false, a[s], false, b0[s], (short)0, acc0, false, false);
            acc1 = __builtin_amdgcn_wmma_f32_16x16x4_f32(false, a[s], false, b1[s], (short)0, acc1, false, false);
            acc2 = __builtin_amdgcn_wmma_f32_16x16x4_f32(false, a[s], false, b2[s], (short)0, acc2, false, false);
            acc3 = __builtin_amdgcn_wmma_f32_16x16x4_f32(false, a[s], false, b3[s], (short)0, acc3, false, false);
            int kn = k0 + 16 + 4 * s + kOff;
            a[s]  = ld(aRow, kn);
            b0[s] = ld(wr0, kn);
            b1[s] = ld(wr1, kn);
            b2[s] = ld(wr2, kn);
            b3[s] = ld(wr3, kn);
        }
    }

    // epilogue: consume last 4 stages (k = K-16 .. K-4), no further loads
    #pragma unroll
    for (int s = 0; s < 4; ++s) {
        acc0 = __builtin_amdgcn_wmma_f32_16x16x4_f32(false, a[s], false, b0[s], (short)0, acc0, false, false);
        acc1 = __builtin_amdgcn_wmma_f32_16x16x4_f32(false, a[s], false, b1[s], (short)0, acc1, false, false);
        acc2 = __builtin_amdgcn_wmma_f32_16x16x4_f32(false, a[s], false, b2[s], (short)0, acc2, false, false);
        acc3 = __builtin_amdgcn_wmma_f32_16x16x4_f32(false, a[s], false, b3[s], (short)0, acc3, false, false);
    }

    // C/D layout: VGPR v -> row mBase+v (lanes 0-15) / mBase+8+v (lanes 16-31); col = lane&15
    const int rowAdd = (lane >> 4) << 3;
    #pragma unroll
    for (int v = 0; v < 8; ++v) {
        int row = mBase + v + rowAdd;
        float vals[4] = { acc0[v], acc1[v], acc2[v], acc3[v] };
        #pragma unroll
        for (int j = 0; j < 4; ++j) {
            int col = nBase + j * 16 + l16;
            float x = vals[j] + bias[col];
            if (GELU) x = 0.5f * x * (1.0f + erff(x * 0.70710678118654752f));
            C[(size_t)row * Ncols + col] = x;
        }
    }
}

extern "C" void kernel_launch(void* const* d_in, const int* in_sizes, int n_in,
                              void* d_out, int out_size, void* d_ws, size_t ws_size,
                              hipStream_t stream) {
    const float* hs     = (const float*)d_in[0];   // (1, 2304, 1280)
    const float* attn   = (const float*)d_in[1];   // (1, 16, 2304, 2304)
    const float* metric = (const float*)d_in[2];   // (1, 2304, 80)
    const float* ln_w   = (const float*)d_in[3];   // (1280,)
    const float* ln_b   = (const float*)d_in[4];   // (1280,)
    const float* w1     = (const float*)d_in[5];   // (5120, 5120)
    const float* b1     = (const float*)d_in[6];   // (5120,)
    const float* w2     = (const float*)d_in[7];   // (3584, 5120)
    const float* b2     = (const float*)d_in[8];   // (3584,)
    float* out = (float*)d_out;                    // (288, 3584) fp32

    // workspace layout (float units; even offsets keep 8B alignment for v2f loads)
    float* ws      = (float*)d_ws;
    float* cls     = ws;                            // 2304
    int*   keep    = (int*)(ws + 2304);             // 2304
    int*   dom_pos = keep + 2304;                   // 976
    int*   filt    = dom_pos + 976;                 // 1344
    float* mnorm   = (float*)(filt + 1344);         // 1332*80 = 106560
    int*   assignv = (int*)(mnorm + 106560);        // 1344
    float* agg     = (float*)(assignv + 1344);      // 180*1280 = 230400
    float* X       = agg + 230400;                  // 1152*1280 = 1474560
    float* Y       = X + 1474560;                   // 288*5120  = 1474560

    k_cls      <<<(NM1 + 255) / 256, 256, 0, stream>>>(attn, cls);
    k_rank     <<<(NM1 + 255) / 256, 256, 0, stream>>>(cls, keep);
    k_positions<<<1, 1, 0, stream>>>(keep, dom_pos, filt);
    k_metric   <<<NF, 128, 0, stream>>>(metric, filt, mnorm);
    k_assign   <<<(NF + 255) / 256, 256, 0, stream>>>(mnorm, assignv);
    k_aggregate<<<CTX, 256, 0, stream>>>(hs, filt, assignv, agg);
    k_final_ln <<<ROWS, 256, 0, stream>>>(hs, dom_pos, agg, ln_w, ln_b, X);

    // GEMM1: X(288x5120) @ w1^T + b1, exact GELU -> Y(288x5120)
    gemm_wmma_f32<true> <<<dim3(XROWS / 16, HS / 256), 128, 0, stream>>>(X, w1, b1, Y, HS, HS);
    // GEMM2: Y(288x5120) @ w2^T + b2 -> out(288x3584)
    gemm_wmma_f32<false><<<dim3(XROWS / 16, N2 / 256), 128, 0, stream>>>(Y, w2, b2, out, HS, N2);
}